// MeanAggregator_55594056680017
// MI455X (gfx1250) — compile-verified
//
#include <hip/hip_runtime.h>

typedef __attribute__((ext_vector_type(2))) float v2f;
typedef __attribute__((ext_vector_type(4))) float v4f;
typedef __attribute__((ext_vector_type(8))) float v8f;

#define NN   16384
#define DD   64
#define KBLK 128
#define KS   132   // LDS stride (dwords): rows 528B -> 16B-aligned for async B128, banks spread
#define KS2  66    // LDS stride for transposed weights (even -> b64 aligned)
#define XTG_BYTES ((size_t)DD * (size_t)NN * 4)

static __device__ __forceinline__ unsigned lds_off(const void* p) {
  // Low 32 bits of a flat LDS address are the in-allocation byte offset.
  return (unsigned)(size_t)p;
}

// ---------------------------------------------------------------------------
// Shared WMMA inner loop over one 128-K chunk: 16x64 tile, fp32 WMMA.
// adj loads are non-temporal (1 GiB single-use stream; keep L2 for X/W/out).
// ---------------------------------------------------------------------------
static __device__ __forceinline__ void wmma_chunk(
    const float* __restrict__ adjp,   // &adj[row_lane*NN + k0 + 2*h]
    const float* __restrict__ xt, int l15, int h,
    v8f& a0, v8f& a1, v8f& a2, v8f& a3, float& rs)
{
  #pragma unroll 4
  for (int kk = 0; kk < KBLK; kk += 4) {
    v2f a = __builtin_nontemporal_load((const v2f*)(adjp + kk));
    rs += a.x + a.y;                  // fused degree accumulation
    const int bb = kk + 2 * h;
    v2f b0 = *(const v2f*)&xt[( 0 + l15) * KS + bb];
    v2f b1 = *(const v2f*)&xt[(16 + l15) * KS + bb];
    v2f b2 = *(const v2f*)&xt[(32 + l15) * KS + bb];
    v2f b3 = *(const v2f*)&xt[(48 + l15) * KS + bb];
    a0 = __builtin_amdgcn_wmma_f32_16x16x4_f32(false, a, false, b0, (short)0, a0, false, false);
    a1 = __builtin_amdgcn_wmma_f32_16x16x4_f32(false, a, false, b1, (short)0, a1, false, false);
    a2 = __builtin_amdgcn_wmma_f32_16x16x4_f32(false, a, false, b2, (short)0, a2, false, false);
    a3 = __builtin_amdgcn_wmma_f32_16x16x4_f32(false, a, false, b3, (short)0, a3, false, false);
  }
}

static __device__ __forceinline__ void neigh_epilogue(
    float* __restrict__ out, long row0, int l15, int h,
    v8f& a0, v8f& a1, v8f& a2, v8f& a3, float rs)
{
  rs += __shfl_xor(rs, 16, 32);       // combine half-wave partial row sums
  float inv[8];
  #pragma unroll
  for (int r = 0; r < 8; ++r) {
    float s = __shfl(rs, r + (h << 3), 32);
    inv[r] = 1.0f / fmaxf(s, 1.0f);
  }
  #pragma unroll
  for (int r = 0; r < 8; ++r) {
    const long o = (row0 + r + 8 * h) * 128 + 64 + l15;
    out[o +  0] = a0[r] * inv[r];
    out[o + 16] = a1[r] * inv[r];
    out[o + 32] = a2[r] * inv[r];
    out[o + 48] = a3[r] * inv[r];
  }
}

// ---------------------------------------------------------------------------
// Kernel 0 (async path only): one-time transpose neighbor_x -> xtg[DD][NN].
// ---------------------------------------------------------------------------
__global__ __launch_bounds__(256) void transpose_x_kernel(
    const float* __restrict__ xnb, float* __restrict__ xtg)
{
  __shared__ float xt[DD * KS];
  const int t  = threadIdx.x;
  const int k0 = blockIdx.x * KBLK;
  {
    const int r = t >> 4, n0 = (t & 15) * 4;
    #pragma unroll
    for (int p = 0; p < 8; ++p) {
      const int kk = p * 16 + r;
      v4f v = *(const v4f*)&xnb[(size_t)(k0 + kk) * DD + n0];
      xt[(n0 + 0) * KS + kk] = v.x;
      xt[(n0 + 1) * KS + kk] = v.y;
      xt[(n0 + 2) * KS + kk] = v.z;
      xt[(n0 + 3) * KS + kk] = v.w;
    }
  }
  __syncthreads();
  {
    const int n = t >> 2, kq = t & 3;
    #pragma unroll
    for (int p = 0; p < 8; ++p) {
      const int j = kq + 4 * p;      // float4 index within the 128-K chunk
      *(v4f*)&xtg[(size_t)n * NN + k0 + j * 4] = *(const v4f*)&xt[n * KS + j * 4];
    }
  }
}

// ---------------------------------------------------------------------------
// Kernel 1a (preferred): double-buffered GLOBAL_LOAD_ASYNC_TO_LDS_B128 staging
// of the X tile from pre-transposed xtg; next tile's DMA overlaps current
// chunk's WMMA work (ASYNCcnt in-order => s_wait_asynccnt 8 = previous group).
// ---------------------------------------------------------------------------
static __device__ __forceinline__ void issue_xtile_async(
    const float* __restrict__ xtg, int k0, int t, unsigned ldsbase)
{
  #pragma unroll
  for (int p = 0; p < 8; ++p) {
    const int i  = p * 256 + t;      // 2048 B128 transfers cover the 32KB tile
    const int n  = i >> 5;           // 0..63
    const int kg = i & 31;           // 0..31 (16B groups along K)
    const unsigned dst = ldsbase + (unsigned)(n * KS * 4 + kg * 16);
    const unsigned long long src =
        (unsigned long long)(size_t)(xtg + (size_t)n * NN + (size_t)k0 + (size_t)kg * 4);
    asm volatile("global_load_async_to_lds_b128 %0, %1, off"
                 :: "v"(dst), "v"(src) : "memory");
  }
}

__global__ __launch_bounds__(256) void neigh_agg_async_kernel(
    const float* __restrict__ adj,
    const float* __restrict__ xtg,
    float* __restrict__ out)
{
  __shared__ float xt[2 * DD * KS];  // double buffer, 67,584 B
  const int  t    = threadIdx.x;
  const int  lane = t & 31;
  const int  wave = t >> 5;
  const int  l15  = lane & 15;
  const int  h    = lane >> 4;
  const long row0 = (long)blockIdx.x * 128 + (long)wave * 16;
  const float* adjp = adj + (row0 + l15) * (long)NN + 2 * h;
  const unsigned lb0 = lds_off(xt);
  const unsigned lb1 = lb0 + DD * KS * 4;

  v8f a0 = {}, a1 = {}, a2 = {}, a3 = {};
  float rs = 0.0f;

  issue_xtile_async(xtg, 0, t, lb0);            // prologue: stage chunk 0
  int k0 = 0;
  for (; k0 + KBLK < NN; k0 += KBLK) {
    const int cur = (k0 / KBLK) & 1;
    issue_xtile_async(xtg, k0 + KBLK, t, cur ? lb0 : lb1);  // prefetch next
    asm volatile("s_wait_asynccnt 0x8" ::: "memory");       // previous 8 done
    __syncthreads();                                        // tile ready (all waves)
    wmma_chunk(adjp + k0, cur ? xt + DD * KS : xt, l15, h, a0, a1, a2, a3, rs);
    __syncthreads();                  // all reads done before this buffer is rewritten
  }
  asm volatile("s_wait_asynccnt 0x0" ::: "memory");         // last tile done
  __syncthreads();
  wmma_chunk(adjp + k0, ((k0 / KBLK) & 1) ? xt + DD * KS : xt,
             l15, h, a0, a1, a2, a3, rs);

  neigh_epilogue(out, row0, l15, h, a0, a1, a2, a3, rs);
}

// ---------------------------------------------------------------------------
// Kernel 1b (fallback if ws_size < 4MB): in-kernel transpose staging.
// ---------------------------------------------------------------------------
__global__ __launch_bounds__(256) void neigh_agg_kernel(
    const float* __restrict__ adj,
    const float* __restrict__ xnb,
    float* __restrict__ out)
{
  __shared__ float xt[DD * KS];
  const int  t    = threadIdx.x;
  const int  lane = t & 31;
  const int  wave = t >> 5;
  const int  l15  = lane & 15;
  const int  h    = lane >> 4;
  const long row0 = (long)blockIdx.x * 128 + (long)wave * 16;
  const float* adjp = adj + (row0 + l15) * (long)NN + 2 * h;

  v8f a0 = {}, a1 = {}, a2 = {}, a3 = {};
  float rs = 0.0f;

  for (int k0 = 0; k0 < NN; k0 += KBLK) {
    __syncthreads();
    {
      const int r = t >> 4, n0 = (t & 15) * 4;
      #pragma unroll
      for (int p = 0; p < 8; ++p) {
        const int kk = p * 16 + r;
        v4f v = *(const v4f*)&xnb[(size_t)(k0 + kk) * DD + n0];
        xt[(n0 + 0) * KS + kk] = v.x;
        xt[(n0 + 1) * KS + kk] = v.y;
        xt[(n0 + 2) * KS + kk] = v.z;
        xt[(n0 + 3) * KS + kk] = v.w;
      }
    }
    __syncthreads();
    wmma_chunk(adjp + k0, xt, l15, h, a0, a1, a2, a3, rs);
  }
  neigh_epilogue(out, row0, l15, h, a0, a1, a2, a3, rs);
}

// ---------------------------------------------------------------------------
// Kernel 2: out[:,0:64] = self_x @ W1 + b1 ; out[:,64:128] = neigh @ W2 + b2
// ---------------------------------------------------------------------------
__global__ __launch_bounds__(256) void linear_kernel(
    const float* __restrict__ sx,
    const float* __restrict__ w1,
    const float* __restrict__ w2,
    const float* __restrict__ b1,
    const float* __restrict__ b2,
    float* __restrict__ out)
{
  __shared__ float wt[2 * DD * KS2];

  const int  t    = threadIdx.x;
  const int  lane = t & 31;
  const int  wave = t >> 5;
  const int  l15  = lane & 15;
  const int  h    = lane >> 4;
  const long row0 = (long)blockIdx.x * 128 + (long)wave * 16;

  {
    const int r = t >> 4, n0 = (t & 15) * 4;
    #pragma unroll
    for (int p = 0; p < 4; ++p) {
      const int k = p * 16 + r;
      v4f a = *(const v4f*)&w1[k * DD + n0];
      v4f b = *(const v4f*)&w2[k * DD + n0];
      #pragma unroll
      for (int j = 0; j < 4; ++j) {
        wt[(n0 + j) * KS2 + k]            = a[j];
        wt[DD * KS2 + (n0 + j) * KS2 + k] = b[j];
      }
    }
  }
  __syncthreads();

  v8f aS0 = {}, aS1 = {}, aS2 = {}, aS3 = {};
  v8f aN0 = {}, aN1 = {}, aN2 = {}, aN3 = {};

  const long srow = (row0 + l15) * (long)DD;
  const long nrow = (row0 + l15) * 128 + 64;

  #pragma unroll 2
  for (int k = 0; k < DD; k += 4) {
    v2f as = *(const v2f*)&sx[srow + k + 2 * h];
    v2f an = *(const v2f*)&out[nrow + k + 2 * h];
    const int bb = k + 2 * h;
    v2f b;
    b = *(const v2f*)&wt[( 0 + l15) * KS2 + bb];
    aS0 = __builtin_amdgcn_wmma_f32_16x16x4_f32(false, as, false, b, (short)0, aS0, false, false);
    b = *(const v2f*)&wt[(16 + l15) * KS2 + bb];
    aS1 = __builtin_amdgcn_wmma_f32_16x16x4_f32(false, as, false, b, (short)0, aS1, false, false);
    b = *(const v2f*)&wt[(32 + l15) * KS2 + bb];
    aS2 = __builtin_amdgcn_wmma_f32_16x16x4_f32(false, as, false, b, (short)0, aS2, false, false);
    b = *(const v2f*)&wt[(48 + l15) * KS2 + bb];
    aS3 = __builtin_amdgcn_wmma_f32_16x16x4_f32(false, as, false, b, (short)0, aS3, false, false);

    b = *(const v2f*)&wt[DD * KS2 + ( 0 + l15) * KS2 + bb];
    aN0 = __builtin_amdgcn_wmma_f32_16x16x4_f32(false, an, false, b, (short)0, aN0, false, false);
    b = *(const v2f*)&wt[DD * KS2 + (16 + l15) * KS2 + bb];
    aN1 = __builtin_amdgcn_wmma_f32_16x16x4_f32(false, an, false, b, (short)0, aN1, false, false);
    b = *(const v2f*)&wt[DD * KS2 + (32 + l15) * KS2 + bb];
    aN2 = __builtin_amdgcn_wmma_f32_16x16x4_f32(false, an, false, b, (short)0, aN2, false, false);
    b = *(const v2f*)&wt[DD * KS2 + (48 + l15) * KS2 + bb];
    aN3 = __builtin_amdgcn_wmma_f32_16x16x4_f32(false, an, false, b, (short)0, aN3, false, false);
  }

  const float bS0 = b1[ 0 + l15], bS1 = b1[16 + l15], bS2 = b1[32 + l15], bS3 = b1[48 + l15];
  const float bN0 = b2[ 0 + l15], bN1 = b2[16 + l15], bN2 = b2[32 + l15], bN3 = b2[48 + l15];

  #pragma unroll
  for (int r = 0; r < 8; ++r) {
    const long o = (row0 + r + 8 * h) * 128 + l15;
    out[o +  0]      = aS0[r] + bS0;
    out[o + 16]      = aS1[r] + bS1;
    out[o + 32]      = aS2[r] + bS2;
    out[o + 48]      = aS3[r] + bS3;
    out[o + 64 +  0] = aN0[r] + bN0;
    out[o + 64 + 16] = aN1[r] + bN1;
    out[o + 64 + 32] = aN2[r] + bN2;
    out[o + 64 + 48] = aN3[r] + bN3;
  }
}

extern "C" void kernel_launch(void* const* d_in, const int* in_sizes, int n_in,
                              void* d_out, int out_size, void* d_ws, size_t ws_size,
                              hipStream_t stream) {
  const float* self_x          = (const float*)d_in[0];
  const float* neighbor_x      = (const float*)d_in[1];
  const float* adj             = (const float*)d_in[2];
  const float* self_weight     = (const float*)d_in[3];
  const float* neighbor_weight = (const float*)d_in[4];
  const float* self_bias       = (const float*)d_in[5];
  const float* neighbor_bias   = (const float*)d_in[6];
  float* out = (float*)d_out;

  dim3 grid(NN / 128), block(256);
  if (ws_size >= XTG_BYTES) {
    float* xtg = (float*)d_ws;
    transpose_x_kernel<<<grid, block, 0, stream>>>(neighbor_x, xtg);
    neigh_agg_async_kernel<<<grid, block, 0, stream>>>(adj, xtg, out);
  } else {
    neigh_agg_kernel<<<grid, block, 0, stream>>>(adj, neighbor_x, out);
  }
  linear_kernel<<<grid, block, 0, stream>>>(self_x, self_weight, neighbor_weight,
                                            self_bias, neighbor_bias, out);
}